// ImprovedBothMamba_55095840473275
// MI455X (gfx1250) — compile-verified
//
#include <hip/hip_runtime.h>
#include <hip/hip_bf16.h>
#include <math.h>

// ---------------------------------------------------------------------------
// CDNA5 / gfx1250 implementation of ImprovedBothMamba forward.
// All dense algebra goes through one batched GEMM kernel built on
// V_WMMA_F32_16X16X4_F32 (wave32). Each wave computes a 32x32 output tile
// (four 16x16 WMMA accumulators; 2 A fragments x 2 B fragments). Fully
// in-bounds tiles take an unguarded b64-load path; ragged tiles take a
// branch-free clamped-address path: loads are issued unconditionally from
// clamped (always-legal) addresses, then zeroed via select (v_cndmask) --
// no exec masking, no branches.
// Workload fits in L2 (192MB) -> fp32 WMMA keeps reference accuracy at no
// bandwidth cost.
// Input flattening assumption: depth-first, dict insertion order:
//   0:x | spa.lsp[1..10] | spa.prca.attn 9x7 [11..73] proj[74,75]
//   spa.mamba[76..84] spa.gn[85,86] | spe.prca[87..151] spe.mamba[152..160]
//   spe.gn[161,162] | bridge[163..168] | ccaf[169..175]
// ---------------------------------------------------------------------------

typedef __attribute__((ext_vector_type(2))) float v2f;
typedef __attribute__((ext_vector_type(8))) float v8f;

static __device__ __forceinline__ float siluf(float v){ return v / (1.f + __expf(-v)); }
static __device__ __forceinline__ float sigf (float v){ return 1.f / (1.f + __expf(-v)); }
static __device__ __forceinline__ int   imin (int a,int b){ return a<b?a:b; }
static __device__ __forceinline__ int   imax (int a,int b){ return a>b?a:b; }

static __device__ __forceinline__ float gemm_epi(float v, int row, int col,
                                                 const float* bias, int biasMode, int act)
{
    if      (biasMode == 1) v += bias[row];
    else if (biasMode == 2) v += bias[col];
    if      (act == 1) v = siluf(v);
    else if (act == 2) v = sigf(v);
    else if (act == 3) v = (v > 20.f) ? v : log1pf(__expf(v));
    return v;
}

#define WMMA_F32(a, b, c) \
    __builtin_amdgcn_wmma_f32_16x16x4_f32(false, (a), false, (b), (short)0, (c), false, false)

// ---------------------------------------------------------------------------
// Batched GEMM:  Out[m,n] = sum_k A[m,k] * B[k,n]  (+bias, +activation)
// transB: B accessed as B[n,k] (row-major weight (N,K))
// biasMode: 0 none, 1 per-row(m), 2 per-col(n).  act: 0 none,1 silu,2 sigmoid,3 softplus
// batch z decomposed as (outer, inner): ptr += zo*s?o + zi*s?i
// Wave tile: 32 rows x 32 cols (four v_wmma_f32_16x16x4_f32 accumulators).
// ---------------------------------------------------------------------------
__global__ void k_gemm(const float* A, const float* Bm, const float* bias, float* O,
                       int M, int N, int K, int lda, int ldb, int ldo,
                       long long sAo, long long sAi, long long sBo, long long sBi,
                       long long sOo, long long sOi, int inner,
                       int transB, int biasMode, int act)
{
    int z  = blockIdx.z;
    int zo = z / inner, zi = z - zo * inner;
    A  += zo * sAo + zi * sAi;
    Bm += zo * sBo + zi * sBi;
    O  += zo * sOo + zi * sOi;

    int n0 = (blockIdx.x * 4 + threadIdx.y) * 32;   // this wave: cols [n0, n0+32)
    if (n0 >= N) return;                             // wave-uniform exit
    int m0 = blockIdx.y * 32;                        // this wave: rows [m0, m0+32)

    int lane = threadIdx.x;                          // 0..31 (wave32)
    int lr = lane & 15;
    int hr = lane >> 4;                              // lane-half -> K pair / M half
    int ar0 = m0 + lr, ar1 = ar0 + 16;
    int bc0 = n0 + lr, bc1 = bc0 + 16;

    v8f acc00 = {}, acc01 = {}, acc10 = {}, acc11 = {};   // [rowTile][colTile]

    bool fast = (m0 + 32 <= M) && (n0 + 32 <= N) && ((K & 3) == 0);
    if (fast) {
        const float* A0 = A + (long long)ar0 * lda + 2 * hr;
        const float* A1 = A + (long long)ar1 * lda + 2 * hr;
        if (transB) {
            const float* B0 = Bm + (long long)bc0 * ldb + 2 * hr;
            const float* B1 = Bm + (long long)bc1 * ldb + 2 * hr;
            #pragma unroll 4
            for (int k0 = 0; k0 < K; k0 += 4) {
                v2f a0 = *(const v2f*)(A0 + k0);
                v2f a1 = *(const v2f*)(A1 + k0);
                v2f b0 = *(const v2f*)(B0 + k0);
                v2f b1 = *(const v2f*)(B1 + k0);
                acc00 = WMMA_F32(a0, b0, acc00);
                acc01 = WMMA_F32(a0, b1, acc01);
                acc10 = WMMA_F32(a1, b0, acc10);
                acc11 = WMMA_F32(a1, b1, acc11);
            }
        } else {
            const float* Bp = Bm + (long long)(2 * hr) * ldb;
            const long long step = 4LL * ldb;
            #pragma unroll 4
            for (int k0 = 0; k0 < K; k0 += 4) {
                v2f a0 = *(const v2f*)(A0 + k0);
                v2f a1 = *(const v2f*)(A1 + k0);
                v2f b0, b1;
                b0.x = Bp[bc0];  b0.y = Bp[(long long)ldb + bc0];
                b1.x = Bp[bc1];  b1.y = Bp[(long long)ldb + bc1];
                Bp += step;
                acc00 = WMMA_F32(a0, b0, acc00);
                acc01 = WMMA_F32(a0, b1, acc01);
                acc10 = WMMA_F32(a1, b0, acc10);
                acc11 = WMMA_F32(a1, b1, acc11);
            }
        }
    } else {
        // Branch-free guarded path: loads are issued UNCONDITIONALLY from
        // clamped (always in-bounds) addresses; zeros applied via select
        // afterwards. No exec masking, no branches in the K loop.
        bool am0 = ar0 < M, am1 = ar1 < M;
        bool bm0 = bc0 < N, bm1 = bc1 < N;
        const float* A0  = A  + (long long)imin(ar0, M - 1) * lda;
        const float* A1  = A  + (long long)imin(ar1, M - 1) * lda;
        const float* B0t = Bm + (long long)imin(bc0, N - 1) * ldb;
        const float* B1t = Bm + (long long)imin(bc1, N - 1) * ldb;
        int c0 = imin(bc0, N - 1), c1 = imin(bc1, N - 1);
        for (int k0 = 0; k0 < K; k0 += 4) {
            int ka = k0 + (hr << 1);
            bool kk0 = ka < K, kk1 = (ka + 1) < K;
            int ka0 = imin(ka, K - 1), ka1 = imin(ka + 1, K - 1);

            float va00 = A0[ka0];
            float va01 = A0[ka1];
            float va10 = A1[ka0];
            float va11 = A1[ka1];
            float vb00, vb01, vb10, vb11;
            if (transB) {
                vb00 = B0t[ka0];  vb01 = B0t[ka1];
                vb10 = B1t[ka0];  vb11 = B1t[ka1];
            } else {
                const float* Br0 = Bm + (long long)ka0 * ldb;
                const float* Br1 = Bm + (long long)ka1 * ldb;
                vb00 = Br0[c0];   vb01 = Br1[c0];
                vb10 = Br0[c1];   vb11 = Br1[c1];
            }

            v2f a0, a1, b0, b1;
            a0.x = (am0 && kk0) ? va00 : 0.f;
            a0.y = (am0 && kk1) ? va01 : 0.f;
            a1.x = (am1 && kk0) ? va10 : 0.f;
            a1.y = (am1 && kk1) ? va11 : 0.f;
            b0.x = (bm0 && kk0) ? vb00 : 0.f;
            b0.y = (bm0 && kk1) ? vb01 : 0.f;
            b1.x = (bm1 && kk0) ? vb10 : 0.f;
            b1.y = (bm1 && kk1) ? vb11 : 0.f;

            acc00 = WMMA_F32(a0, b0, acc00);
            acc01 = WMMA_F32(a0, b1, acc01);
            acc10 = WMMA_F32(a1, b0, acc10);
            acc11 = WMMA_F32(a1, b1, acc11);
        }
    }

    // ---- epilogue ----
    if (fast) {
        #pragma unroll
        for (int r = 0; r < 8; ++r) {
            int or0 = m0 + r + (hr << 3);
            int or1 = or0 + 16;
            O[(long long)or0 * ldo + bc0] = gemm_epi(acc00[r], or0, bc0, bias, biasMode, act);
            O[(long long)or0 * ldo + bc1] = gemm_epi(acc01[r], or0, bc1, bias, biasMode, act);
            O[(long long)or1 * ldo + bc0] = gemm_epi(acc10[r], or1, bc0, bias, biasMode, act);
            O[(long long)or1 * ldo + bc1] = gemm_epi(acc11[r], or1, bc1, bias, biasMode, act);
        }
    } else {
        #pragma unroll
        for (int r = 0; r < 8; ++r) {
            int or0 = m0 + r + (hr << 3);
            int or1 = or0 + 16;
            if (or0 < M) {
                if (bc0 < N)
                    O[(long long)or0 * ldo + bc0] =
                        gemm_epi(acc00[r], or0, bc0, bias, biasMode, act);
                if (bc1 < N)
                    O[(long long)or0 * ldo + bc1] =
                        gemm_epi(acc01[r], or0, bc1, bias, biasMode, act);
            }
            if (or1 < M) {
                if (bc0 < N)
                    O[(long long)or1 * ldo + bc0] =
                        gemm_epi(acc10[r], or1, bc0, bias, biasMode, act);
                if (bc1 < N)
                    O[(long long)or1 * ldo + bc1] =
                        gemm_epi(acc11[r], or1, bc1, bias, biasMode, act);
            }
        }
    }
}

// ---------------------------------------------------------------------------
// Depthwise 3x3 conv (padding = dilation), NCHW, groups=C
// ---------------------------------------------------------------------------
__global__ void k_dwconv3x3(const float* __restrict__ x, const float* __restrict__ w,
                            const float* __restrict__ bias, float* __restrict__ out,
                            int Bb, int C, int H, int W, int dil)
{
    long long idx = blockIdx.x * (long long)blockDim.x + threadIdx.x;
    long long total = (long long)Bb * C * H * W;
    if (idx >= total) return;
    int wx = (int)(idx % W);
    int hy = (int)((idx / W) % H);
    int c  = (int)((idx / ((long long)W * H)) % C);
    int b  = (int)( idx / ((long long)W * H * C));
    const float* wc = w + c * 9;
    const float* xb = x + (((long long)b * C + c) * H) * W;
    float s = bias[c];
    #pragma unroll
    for (int i = 0; i < 3; ++i)
        #pragma unroll
        for (int j = 0; j < 3; ++j) {
            int yy = hy + dil * (i - 1), xx = wx + dil * (j - 1);
            if (yy >= 0 && yy < H && xx >= 0 && xx < W)
                s += wc[i * 3 + j] * xb[(long long)yy * W + xx];
        }
    out[idx] = s;
}

// ---------------------------------------------------------------------------
// GroupNorm (+silu, +residual):  out = act(gn(x)) + res ; one block per (b,g)
// ---------------------------------------------------------------------------
__global__ void k_groupnorm(const float* __restrict__ x, const float* __restrict__ gw,
                            const float* __restrict__ gb, const float* __restrict__ res,
                            float* __restrict__ out, int C, long long HW, int G, int doSilu)
{
    int b = blockIdx.x / G, g = blockIdx.x % G;
    int cg = C / G;
    long long base = ((long long)b * C + (long long)g * cg) * HW;
    long long n = (long long)cg * HW;
    __shared__ float sm[256], sv[256];
    float ls = 0.f, lq = 0.f;
    for (long long i = threadIdx.x; i < n; i += blockDim.x) {
        float v = x[base + i]; ls += v; lq += v * v;
    }
    sm[threadIdx.x] = ls; sv[threadIdx.x] = lq;
    __syncthreads();
    for (int st = blockDim.x / 2; st > 0; st >>= 1) {
        if ((int)threadIdx.x < st) { sm[threadIdx.x] += sm[threadIdx.x + st];
                                     sv[threadIdx.x] += sv[threadIdx.x + st]; }
        __syncthreads();
    }
    float mean = sm[0] / (float)n;
    float var  = sv[0] / (float)n - mean * mean;
    float rstd = rsqrtf(var + 1e-5f);
    for (long long i = threadIdx.x; i < n; i += blockDim.x) {
        int c = g * cg + (int)(i / HW);
        float v = (x[base + i] - mean) * rstd * gw[c] + gb[c];
        if (doSilu) v = siluf(v);
        if (res) v += res[base + i];
        out[base + i] = v;
    }
}

// L2-normalize one channel row of the q or k region of the qkv buffer
__global__ void k_rownorm(float* qkv, int C, int C3, int coff, int Ns)
{
    int b = blockIdx.x / C, c = blockIdx.x % C;
    float* row = qkv + ((long long)b * C3 + coff + c) * Ns;
    __shared__ float s[256];
    float l = 0.f;
    for (int i = threadIdx.x; i < Ns; i += blockDim.x) { float v = row[i]; l += v * v; }
    s[threadIdx.x] = l; __syncthreads();
    for (int st = blockDim.x / 2; st > 0; st >>= 1) {
        if ((int)threadIdx.x < st) s[threadIdx.x] += s[threadIdx.x + st];
        __syncthreads();
    }
    float inv = 1.f / fmaxf(sqrtf(s[0]), 1e-12f);
    for (int i = threadIdx.x; i < Ns; i += blockDim.x) row[i] *= inv;
}

// softmax over rows of (B*heads, 32, 32) after scaling by temp[head]; 32 lanes/row
__global__ void k_softmax32(float* a, const float* temp, int heads)
{
    int mat = blockIdx.x >> 5, r = blockIdx.x & 31;
    int h = mat % heads;
    float* row = a + ((long long)mat * 32 + r) * 32;
    float v = row[threadIdx.x] * temp[h];
    float m = v;
    for (int o = 16; o > 0; o >>= 1) m = fmaxf(m, __shfl_xor(m, o));
    float e = __expf(v - m);
    float s = e;
    for (int o = 16; o > 0; o >>= 1) s += __shfl_xor(s, o);
    row[threadIdx.x] = e / s;
}

__global__ void k_avgpool(const float* __restrict__ x, float* __restrict__ o,
                          int C, int H, int W, int f, long long total)
{
    long long idx = blockIdx.x * (long long)blockDim.x + threadIdx.x;
    if (idx >= total) return;
    int Ho = H / f, Wo = W / f;
    int wo = (int)(idx % Wo);
    int ho = (int)((idx / Wo) % Ho);
    int c  = (int)((idx / ((long long)Wo * Ho)) % C);
    int b  = (int)( idx / ((long long)Wo * Ho * C));
    const float* p = x + (((long long)b * C + c) * H) * W;
    float s = 0.f;
    for (int i = 0; i < f; ++i)
        for (int j = 0; j < f; ++j)
            s += p[(long long)(ho * f + i) * W + wo * f + j];
    o[idx] = s / (float)(f * f);
}

// bilinear upsample (half-pixel) writing into concat slot [oc0, oc0+C) of (B, OC, Ho*Wo)
__global__ void k_upsample(const float* __restrict__ x, float* __restrict__ dst,
                           int C, int Hi, int Wi, int Ho, int Wo, int OC, int oc0,
                           long long total)
{
    long long idx = blockIdx.x * (long long)blockDim.x + threadIdx.x;
    if (idx >= total) return;
    int wo = (int)(idx % Wo);
    int ho = (int)((idx / Wo) % Ho);
    int c  = (int)((idx / ((long long)Wo * Ho)) % C);
    int b  = (int)( idx / ((long long)Wo * Ho * C));
    float fy = (ho + 0.5f) * (float)Hi / Ho - 0.5f;
    float fx = (wo + 0.5f) * (float)Wi / Wo - 0.5f;
    int y0 = (int)floorf(fy), x0 = (int)floorf(fx);
    float wy = fy - y0, wx = fx - x0;
    int y0c = imin(imax(y0, 0), Hi - 1), y1c = imin(imax(y0 + 1, 0), Hi - 1);
    int x0c = imin(imax(x0, 0), Wi - 1), x1c = imin(imax(x0 + 1, 0), Wi - 1);
    const float* p = x + ((long long)b * C + c) * Hi * Wi;
    float v = (1.f - wy) * ((1.f - wx) * p[(long long)y0c * Wi + x0c] + wx * p[(long long)y0c * Wi + x1c])
            +         wy * ((1.f - wx) * p[(long long)y1c * Wi + x0c] + wx * p[(long long)y1c * Wi + x1c]);
    dst[((long long)b * OC + oc0 + c) * ((long long)Ho * Wo) + (long long)ho * Wo + wo] = v;
}

// copy (B,C,N) into concat slot oc0 of (B,OC,N)
__global__ void k_copy_slot(const float* __restrict__ src, float* __restrict__ dst,
                            int C, int N, int OC, int oc0, long long total)
{
    long long idx = blockIdx.x * (long long)blockDim.x + threadIdx.x;
    if (idx >= total) return;
    int n = (int)(idx % N);
    int c = (int)((idx / N) % C);
    int b = (int)( idx / ((long long)N * C));
    dst[((long long)b * OC + oc0 + c) * N + n] = src[idx];
}

// (B,C,N) -> (B,N,C)
__global__ void k_transpose_cn(const float* __restrict__ x, float* __restrict__ o,
                               int C, int N, long long total)
{
    long long idx = blockIdx.x * (long long)blockDim.x + threadIdx.x;
    if (idx >= total) return;
    int n = (int)(idx % N);
    int c = (int)((idx / N) % C);
    int b = (int)( idx / ((long long)N * C));
    o[((long long)b * N + n) * C + c] = x[idx];
}

// (B,N,C) -> (B,C,N)
__global__ void k_transpose_nc(const float* __restrict__ x, float* __restrict__ o,
                               int C, int N, long long total)
{
    long long idx = blockIdx.x * (long long)blockDim.x + threadIdx.x;
    if (idx >= total) return;
    int n = (int)(idx % N);
    int c = (int)((idx / N) % C);
    int b = (int)( idx / ((long long)N * C));
    o[idx] = x[((long long)b * N + n) * C + c];
}

// elementwise: op 0 mul, 1 add, 2 absdiff
__global__ void k_ew(const float* __restrict__ a, const float* __restrict__ b,
                     float* __restrict__ o, long long n, int op)
{
    long long idx = blockIdx.x * (long long)blockDim.x + threadIdx.x;
    if (idx >= n) return;
    float x = a[idx], y = b[idx];
    o[idx] = (op == 0) ? x * y : (op == 1) ? x + y : fabsf(x - y);
}

// out[b,c,n] = a[b,c,n] * g[b,n]   (gate (B,1,HW) broadcast over C)
__global__ void k_mulgate(const float* __restrict__ a, const float* __restrict__ g,
                          float* __restrict__ o, int C, long long HW, long long total)
{
    long long idx = blockIdx.x * (long long)blockDim.x + threadIdx.x;
    if (idx >= total) return;
    long long n = idx % HW;
    long long b = idx / (HW * C);
    o[idx] = a[idx] * g[b * HW + n];
}

// x *= (1 + gamma * gate[b,c])
__global__ void k_bridgeapply(float* __restrict__ x, const float* __restrict__ gate,
                              const float* __restrict__ gamma, int C, long long HW,
                              long long total)
{
    long long idx = blockIdx.x * (long long)blockDim.x + threadIdx.x;
    if (idx >= total) return;
    int c = (int)((idx / HW) % C);
    long long b = idx / (HW * C);
    x[idx] *= (1.f + gamma[0] * gate[b * C + c]);
}

// global average pool: one block per (b,c) row of length HW
__global__ void k_gap(const float* __restrict__ x, float* __restrict__ o, int HW)
{
    long long base = (long long)blockIdx.x * HW;
    __shared__ float s[256];
    float l = 0.f;
    for (int i = threadIdx.x; i < HW; i += blockDim.x) l += x[base + i];
    s[threadIdx.x] = l; __syncthreads();
    for (int st = blockDim.x / 2; st > 0; st >>= 1) {
        if ((int)threadIdx.x < st) s[threadIdx.x] += s[threadIdx.x + st];
        __syncthreads();
    }
    if (threadIdx.x == 0) o[blockIdx.x] = s[0] / (float)HW;
}

// Mamba causal depthwise conv1d (k=4) + bias + silu. xz layout (S*L, ld), xin = cols [0,Di)
__global__ void k_mamba_conv(const float* __restrict__ xz, const float* __restrict__ w,
                             const float* __restrict__ bias, float* __restrict__ xc,
                             int S, int L, int Di, int ld, long long total)
{
    long long idx = blockIdx.x * (long long)blockDim.x + threadIdx.x;
    if (idx >= total) return;
    int d = (int)(idx % Di);
    int l = (int)((idx / Di) % L);
    int s = (int)( idx / ((long long)Di * L));
    float acc = bias[d];
    #pragma unroll
    for (int j = 0; j < 4; ++j) {
        int lj = l - 3 + j;
        if (lj >= 0) acc += w[d * 4 + j] * xz[((long long)s * L + lj) * ld + d];
    }
    xc[idx] = siluf(acc);
}

// Selective scan. thread = (s,d). N-states = 16. y = (scan + xc*D) * silu(z)
__global__ void k_mamba_scan(const float* __restrict__ dt, const float* __restrict__ xc,
                             const float* __restrict__ proj, const float* __restrict__ A_log,
                             const float* __restrict__ Dp, const float* __restrict__ xz,
                             float* __restrict__ y, int S, int L, int Di, int R,
                             int projLd, int zLd)
{
    long long idx = blockIdx.x * (long long)blockDim.x + threadIdx.x;
    if (idx >= (long long)S * Di) return;
    int d = (int)(idx % Di);
    int s = (int)(idx / Di);
    float A[16], h[16];
    #pragma unroll
    for (int n = 0; n < 16; ++n) { A[n] = -__expf(A_log[d * 16 + n]); h[n] = 0.f; }
    for (int l = 0; l < L; ++l) {
        long long r = (long long)s * L + l;
        float dtv = dt[r * Di + d];
        float xv  = xc[r * Di + d];
        const float* pr = proj + r * projLd;
        float yv = 0.f;
        #pragma unroll
        for (int n = 0; n < 16; ++n) {
            float dA = __expf(dtv * A[n]);
            h[n] = dA * h[n] + (dtv * xv) * pr[R + n];
            yv  += h[n] * pr[R + 16 + n];
        }
        yv += xv * Dp[d];
        float zv = xz[r * zLd + Di + d];
        y[r * Di + d] = yv * siluf(zv);
    }
}

// final CCAF combine
__global__ void k_ccaf_final(const float* __restrict__ spa, const float* __restrict__ spe,
                             const float* __restrict__ l0, const float* __restrict__ l1,
                             const float* __restrict__ gcons, const float* __restrict__ gconf,
                             const float* __restrict__ beta, float* __restrict__ out,
                             int C, long long HW, long long total)
{
    long long idx = blockIdx.x * (long long)blockDim.x + threadIdx.x;
    if (idx >= total) return;
    int c = (int)((idx / HW) % C);
    long long b = idx / (HW * C);
    float a = spa[idx], e = spe[idx];
    float w0 = sigf(l0[b * C + c] - l1[b * C + c]);   // softmax over {spa,spe}
    float comp = w0 * a + (1.f - w0) * e;
    float cons = gcons[b * C + c] * 0.5f * (a + e);
    out[idx] = comp + beta[0] * cons * (1.f - gconf[b * C + c]);
}

// ---------------------------------------------------------------------------
// Host side
// ---------------------------------------------------------------------------
static inline int cdiv64(long long a, long long b) { return (int)((a + b - 1) / b); }

static void gemmL(hipStream_t st, const float* A, const float* Bm, const float* bias, float* O,
                  int M, int N, int K, int lda, int ldb, int ldo,
                  long long sAo, long long sAi, long long sBo, long long sBi,
                  long long sOo, long long sOi, int inner, int batch,
                  int transB, int biasMode, int act)
{
    dim3 blk(32, 4, 1);
    dim3 grd((unsigned)cdiv64(N, 128), (unsigned)cdiv64(M, 32), (unsigned)batch);
    k_gemm<<<grd, blk, 0, st>>>(A, Bm, bias, O, M, N, K, lda, ldb, ldo,
                                sAo, sAi, sBo, sBi, sOo, sOi, inner, transB, biasMode, act);
}

extern "C" void kernel_launch(void* const* d_in, const int* in_sizes, int n_in,
                              void* d_out, int out_size, void* d_ws, size_t ws_size,
                              hipStream_t stream)
{
    (void)in_sizes; (void)n_in; (void)out_size; (void)ws_size;
    const int B = 2, C = 128, H = 48, W = 48, heads = 4, hd = 32;
    const long long HW = (long long)H * W;      // 2304
    const long long totC = (long long)B * C * HW;

    auto Fp = [&](int i) -> const float* { return (const float*)d_in[i]; };

    // ---- workspace bump allocator ----
    char* wsb = (char*)d_ws;
    size_t off = 0;
    auto alloc = [&](size_t nf) -> float* {
        float* p = (float*)(wsb + off);
        off += ((nf * sizeof(float) + 255) / 256) * 256;
        return p;
    };
    float* xp    = alloc(totC);
    float* xr    = alloc(totC);
    float* t3a   = alloc((size_t)B * 3 * C * HW);
    float* t3b   = alloc((size_t)B * 3 * C * HW);
    float* to1   = alloc(totC);
    float* to2   = alloc(totC);
    float* outs0 = alloc(totC);
    float* outs1 = alloc(totC);
    float* outs2 = alloc(totC);
    float* outs[3] = { outs0, outs1, outs2 };
    float* cat   = alloc((size_t)B * 3 * C * HW);
    float* attnb = alloc((size_t)B * heads * hd * hd);
    float* gbuf  = alloc((size_t)B * HW);
    float* seqin = alloc(totC);
    float* xzb   = alloc((size_t)2359296);   // max(4608*512, 18432*128)
    float* xcb   = alloc((size_t)1179648);   // max(4608*256, 18432*64)
    float* projb = alloc((size_t)626688);    // max(4608*40, 18432*34)
    float* dtb   = alloc((size_t)1179648);
    float* yb    = alloc((size_t)1179648);
    float* seqout= alloc(totC);
    float* spa_o = alloc(totC);
    float* spe_o = alloc(totC);
    float* tmp1  = alloc(totC);
    float* gapbuf= alloc(256);
    float* gsmall= alloc(64);
    float* gateA = alloc(256);
    float* gateB = alloc(256);
    float* l0b   = alloc(256);
    float* l1b   = alloc(256);
    float* gconsb= alloc(256);
    float* gconfb= alloc(256);

    const float* x = Fp(0);

    // one pyramid-attention layer: in -> out (uses t3a, t3b, to1, attnb)
    auto pyr = [&](const float* inb, float* outb, int lb, int Hs, int Ws) {
        int C3 = 3 * C;
        long long Ns = (long long)Hs * Ws;
        // qkv 1x1 conv: W(3C,C) x in(C,Ns) per batch
        gemmL(stream, Fp(lb + 0), inb, Fp(lb + 1), t3a, C3, (int)Ns, C, C, (int)Ns, (int)Ns,
              0, 0, (long long)C * Ns, 0, (long long)C3 * Ns, 0, 1, B, 0, 1, 0);
        { long long tot = (long long)B * C3 * Ns;
          k_dwconv3x3<<<cdiv64(tot, 256), 256, 0, stream>>>(t3a, Fp(lb + 2), Fp(lb + 3),
                                                            t3b, B, C3, Hs, Ws, 2); }
        k_rownorm<<<B * C, 256, 0, stream>>>(t3b, C, C3, 0, (int)Ns);   // q
        k_rownorm<<<B * C, 256, 0, stream>>>(t3b, C, C3, C, (int)Ns);   // k
        // attn = q @ k^T per (b,h): M=N=hd, K=Ns
        gemmL(stream, t3b, t3b + (long long)C * Ns, nullptr, attnb,
              hd, hd, (int)Ns, (int)Ns, (int)Ns, hd,
              (long long)C3 * Ns, (long long)hd * Ns,
              (long long)C3 * Ns, (long long)hd * Ns,
              (long long)heads * hd * hd, (long long)hd * hd,
              heads, B * heads, 1, 0, 0);
        k_softmax32<<<B * heads * hd, 32, 0, stream>>>(attnb, Fp(lb + 4), heads);
        // out = attn @ v
        gemmL(stream, attnb, t3b + (long long)2 * C * Ns, nullptr, to1,
              hd, (int)Ns, hd, hd, (int)Ns, (int)Ns,
              (long long)heads * hd * hd, (long long)hd * hd,
              (long long)C3 * Ns, (long long)hd * Ns,
              (long long)C * Ns, (long long)hd * Ns,
              heads, B * heads, 0, 0, 0);
        // proj 1x1
        gemmL(stream, Fp(lb + 5), to1, Fp(lb + 6), outb, C, (int)Ns, C, C, (int)Ns, (int)Ns,
              0, 0, (long long)C * Ns, 0, (long long)C * Ns, 0, 1, B, 0, 1, 0);
    };

    auto run_prca = [&](const float* xin, float* xr_out, int pb) {
        for (int sc = 0; sc < 3; ++sc) {
            int f = 1 << sc, Hs = H / f, Ws = W / f;
            long long Ns = (long long)Hs * Ws;
            const float* cur;
            if (sc == 0) cur = xin;
            else {
                long long tot = (long long)B * C * Ns;
                k_avgpool<<<cdiv64(tot, 256), 256, 0, stream>>>(xin, to2, C, H, W, f, tot);
                cur = to2;
            }
            for (int ly = 0; ly < 3; ++ly) {
                int lb = pb + sc * 21 + ly * 7;
                const float* src = (ly == 0) ? cur : ((ly == 1) ? outs[sc] : to2);
                float* dst = (ly == 1) ? to2 : outs[sc];
                pyr(src, dst, lb, Hs, Ws);
            }
            if (sc == 0) {
                long long tot = (long long)B * C * Ns;
                k_copy_slot<<<cdiv64(tot, 256), 256, 0, stream>>>(outs[0], cat, C, (int)Ns,
                                                                  3 * C, 0, tot);
            } else {
                long long tot = totC;
                k_upsample<<<cdiv64(tot, 256), 256, 0, stream>>>(outs[sc], cat, C, Hs, Ws,
                                                                 H, W, 3 * C, sc * C, tot);
            }
        }
        // final proj: W(C,3C) x cat(3C,HW)
        gemmL(stream, Fp(pb + 63), cat, Fp(pb + 64), xr_out, C, (int)HW, 3 * C,
              3 * C, (int)HW, (int)HW,
              0, 0, (long long)3 * C * HW, 0, (long long)C * HW, 0, 1, B, 0, 1, 0);
    };

    auto run_mamba = [&](int mb, int S, int L, int D, int Di, int R,
                         const float* sin_, float* sout_) {
        int rows = S * L, PN = R + 32;
        gemmL(stream, sin_, Fp(mb + 0), nullptr, xzb, rows, 2 * Di, D, D, D, 2 * Di,
              0, 0, 0, 0, 0, 0, 1, 1, 1, 0, 0);
        { long long tot = (long long)rows * Di;
          k_mamba_conv<<<cdiv64(tot, 256), 256, 0, stream>>>(xzb, Fp(mb + 1), Fp(mb + 2),
                                                             xcb, S, L, Di, 2 * Di, tot); }
        gemmL(stream, xcb, Fp(mb + 3), nullptr, projb, rows, PN, Di, Di, Di, PN,
              0, 0, 0, 0, 0, 0, 1, 1, 1, 0, 0);
        gemmL(stream, projb, Fp(mb + 4), Fp(mb + 5), dtb, rows, Di, R, PN, R, Di,
              0, 0, 0, 0, 0, 0, 1, 1, 1, 2, 3);
        k_mamba_scan<<<cdiv64((long long)S * Di, 256), 256, 0, stream>>>(
            dtb, xcb, projb, Fp(mb + 6), Fp(mb + 7), xzb, yb, S, L, Di, R, PN, 2 * Di);
        gemmL(stream, yb, Fp(mb + 8), nullptr, sout_, rows, D, Di, Di, Di, D,
              0, 0, 0, 0, 0, 0, 1, 1, 1, 0, 0);
    };

    auto run_gate = [&](const float* xin, const float* w1, int Fh, const float* w2, float* gout) {
        k_gap<<<B * C, 256, 0, stream>>>(xin, gapbuf, (int)HW);
        gemmL(stream, gapbuf, w1, nullptr, gsmall, B, Fh, C, C, C, Fh,
              0, 0, 0, 0, 0, 0, 1, 1, 1, 0, 1);               // silu
        gemmL(stream, gsmall, w2, nullptr, gout, B, C, Fh, Fh, Fh, C,
              0, 0, 0, 0, 0, 0, 1, 1, 1, 0, 2);               // sigmoid
    };

    // ===================== SPA branch =====================
    // LSP
    k_dwconv3x3<<<cdiv64(totC, 256), 256, 0, stream>>>(x, Fp(1), Fp(2), to1, B, C, H, W, 1);
    gemmL(stream, Fp(3), x, Fp(4), to2, 32, (int)HW, C, C, (int)HW, (int)HW,
          0, 0, (long long)C * HW, 0, (long long)32 * HW, 0, 1, B, 0, 1, 1);      // g1 + silu
    gemmL(stream, Fp(5), to2, Fp(6), gbuf, 1, (int)HW, 32, 32, (int)HW, (int)HW,
          0, 0, (long long)32 * HW, 0, HW, 0, 1, B, 0, 1, 2);                     // g2 + sigmoid
    k_mulgate<<<cdiv64(totC, 256), 256, 0, stream>>>(to1, gbuf, tmp1, C, HW, totC);
    gemmL(stream, Fp(7), tmp1, Fp(8), to2, C, (int)HW, C, C, (int)HW, (int)HW,
          0, 0, (long long)C * HW, 0, (long long)C * HW, 0, 1, B, 0, 1, 0);       // pw conv
    k_groupnorm<<<B * 4, 256, 0, stream>>>(to2, Fp(9), Fp(10), x, xp, C, HW, 4, 1);

    run_prca(xp, xr, 11);
    k_transpose_cn<<<cdiv64(totC, 256), 256, 0, stream>>>(xr, seqin, C, (int)HW, totC);
    run_mamba(76, B, (int)HW, C, 256, 8, seqin, seqout);
    k_transpose_nc<<<cdiv64(totC, 256), 256, 0, stream>>>(seqout, to1, C, (int)HW, totC);
    k_groupnorm<<<B * 4, 256, 0, stream>>>(to1, Fp(85), Fp(86), xp, spa_o, C, HW, 4, 1);

    // ===================== SPE branch =====================
    run_prca(x, xr, 87);
    k_transpose_cn<<<cdiv64(totC, 256), 256, 0, stream>>>(xr, seqin, C, (int)HW, totC);
    run_mamba(152, B * (int)HW, 4, 32, 64, 2, seqin, seqout);   // (B*HW) seqs of len 4, D=32
    k_transpose_nc<<<cdiv64(totC, 256), 256, 0, stream>>>(seqout, to1, C, (int)HW, totC);
    k_groupnorm<<<B * 4, 256, 0, stream>>>(to1, Fp(161), Fp(162), x, spe_o, C, HW, 4, 1);

    // ===================== Bridge =====================
    run_gate(spe_o, Fp(163), 32, Fp(164), gateA);   // gate from spe, applied to spa
    run_gate(spa_o, Fp(165), 32, Fp(166), gateB);   // gate from spa, applied to spe
    k_bridgeapply<<<cdiv64(totC, 256), 256, 0, stream>>>(spa_o, gateA, Fp(167), C, HW, totC);
    k_bridgeapply<<<cdiv64(totC, 256), 256, 0, stream>>>(spe_o, gateB, Fp(168), C, HW, totC);

    // ===================== CCAF =====================
    k_gap<<<B * C, 256, 0, stream>>>(spa_o, gapbuf, (int)HW);
    gemmL(stream, gapbuf, Fp(169), nullptr, l0b, B, C, C, C, C, C,
          0, 0, 0, 0, 0, 0, 1, 1, 1, 0, 0);
    k_gap<<<B * C, 256, 0, stream>>>(spe_o, gapbuf, (int)HW);
    gemmL(stream, gapbuf, Fp(170), nullptr, l1b, B, C, C, C, C, C,
          0, 0, 0, 0, 0, 0, 1, 1, 1, 0, 0);
    k_ew<<<cdiv64(totC, 256), 256, 0, stream>>>(spa_o, spe_o, tmp1, totC, 0);   // spa*spe
    run_gate(tmp1, Fp(171), 16, Fp(172), gconsb);
    k_ew<<<cdiv64(totC, 256), 256, 0, stream>>>(spa_o, spe_o, tmp1, totC, 2);   // |spa-spe|
    run_gate(tmp1, Fp(173), 16, Fp(174), gconfb);
    k_ccaf_final<<<cdiv64(totC, 256), 256, 0, stream>>>(spa_o, spe_o, l0b, l1b,
                                                        gconsb, gconfb, Fp(175),
                                                        (float*)d_out, C, HW, totC);
}